// MultiClassCostSensitiveLoss_22522808500489
// MI455X (gfx1250) — compile-verified
//
#include <hip/hip_runtime.h>
#include <hip/hip_bf16.h>

// MultiClassCostSensitiveLoss for MI455X (gfx1250, wave32).
// Memory-bound streaming reduction: ~134 MB traffic -> ~5.8 us at 23.3 TB/s.
// Stage 1: grid-stride float4/int4 streaming, per-thread f32 accum,
//          wave32 shfl_xor tree + LDS -> 1024 deterministic block partials.
// Stage 2: single wave reduces 1024 partials exactly with
//          V_WMMA_F32_16X16X4_F32 (A = partials, B = all-ones, C accum).

typedef float v2f __attribute__((ext_vector_type(2)));
typedef float v8f __attribute__((ext_vector_type(8)));

#define RED_BLOCKS 1024
#define RED_THREADS 256

__device__ __forceinline__ float sample_loss(float a, float b, float c, int t) {
    // t in {-1,0,1}; shifted class idx = t+1 selects logit s, others o1,o2.
    // t=-1: s=a,o1=b,o2=c ; t=0: s=b,o1=a,o2=c ; t=1: s=c,o1=a,o2=b
    float s  = (t < 0) ? a : ((t == 0) ? b : c);
    float o1 = (t < 0) ? b : a;
    float o2 = (t > 0) ? b : c;
    // softmax gather: p = e^s / (e^s+e^o1+e^o2) = 1/(1 + e^(o1-s) + e^(o2-s))
    float denom = 1.0f + __expf(o1 - s) + __expf(o2 - s);
    float p = __builtin_amdgcn_rcpf(denom);   // v_rcp_f32, ~1 ulp
    float d = p - (float)t;
    float w = (t == 0) ? 0.5f : 2.0f;         // weights {2.0, 0.5, 2.0} by idx
    return w * d * d;
}

__global__ void mcc_loss_partials(const float* __restrict__ yp,
                                  const int*   __restrict__ yt,
                                  float* __restrict__ ws, int n) {
    const int tid      = blockIdx.x * blockDim.x + threadIdx.x;
    const int nthreads = gridDim.x * blockDim.x;
    const int ngroups  = n >> 2;              // 4 samples (48 B logits) per group

    const float4* __restrict__ yp4 = reinterpret_cast<const float4*>(yp);
    const int4*   __restrict__ yt4 = reinterpret_cast<const int4*>(yt);

    float acc = 0.0f;
    for (int g = tid; g < ngroups; g += nthreads) {
        float4 p0 = yp4[3 * g + 0];           // 3x128b loads: 4 samples' logits
        float4 p1 = yp4[3 * g + 1];
        float4 p2 = yp4[3 * g + 2];
        int4   t  = yt4[g];                   // 128b load: 4 labels
        acc += sample_loss(p0.x, p0.y, p0.z, t.x);
        acc += sample_loss(p0.w, p1.x, p1.y, t.y);
        acc += sample_loss(p1.z, p1.w, p2.x, t.z);
        acc += sample_loss(p2.y, p2.z, p2.w, t.w);
    }
    // tail (N is 2^23 so normally empty; kept for generality)
    for (int i = (ngroups << 2) + tid; i < n; i += nthreads) {
        acc += sample_loss(yp[3 * i], yp[3 * i + 1], yp[3 * i + 2], yt[i]);
    }

    // wave32 butterfly reduction
    #pragma unroll
    for (int off = 16; off >= 1; off >>= 1)
        acc += __shfl_xor(acc, off, 32);

    __shared__ float lds[RED_THREADS / 32];
    const int lane = threadIdx.x & 31;
    const int wave = threadIdx.x >> 5;
    if (lane == 0) lds[wave] = acc;
    __syncthreads();
    if (threadIdx.x == 0) {
        float s = 0.0f;
        #pragma unroll
        for (int w = 0; w < RED_THREADS / 32; ++w) s += lds[w];
        ws[blockIdx.x] = s;                   // deterministic partial per block
    }
}

// One wave (32 threads, EXEC all-ones). Reduces npart (multiple of 64) floats
// exactly with f32 WMMA: D = A(16x4) * ones(4x16) + C -> C[m][n] = rowsum[m].
// A layout (ISA 7.12.2): lanes 0-15 hold A[lane][0..1] in v0,v1; lanes 16-31
// hold A[lane-16][2..3]. All-ones B is layout-invariant, so any assignment of
// the 64 values into A sums correctly.
__global__ void mcc_loss_final_wmma(const float* __restrict__ ws,
                                    float* __restrict__ out, int npart) {
    const int lane = threadIdx.x;             // 0..31
    v8f c = {};
    v2f bones;
    bones[0] = 1.0f;
    bones[1] = 1.0f;

    for (int base = 0; base < npart; base += 64) {
        v2f a;
        a[0] = ws[base + lane];
        a[1] = ws[base + 32 + lane];
        // 8 args: (neg_a, A, neg_b, B, c_mod, C, reuse_a, reuse_b)
        c = __builtin_amdgcn_wmma_f32_16x16x4_f32(
                false, a, false, bones, (short)0, c, false, false);
    }

    // C/D layout: lane 0 VGPR r = D[r][0] = rowsum[r] (r=0..7),
    //             lane 16 VGPR r = D[8+r][0] = rowsum[8+r].
    float s = c[0] + c[1] + c[2] + c[3] + c[4] + c[5] + c[6] + c[7];
    float hi = __shfl(s, 16, 32);
    if (lane == 0) out[0] = s + hi;
}

extern "C" void kernel_launch(void* const* d_in, const int* in_sizes, int n_in,
                              void* d_out, int out_size, void* d_ws, size_t ws_size,
                              hipStream_t stream) {
    const float* yp  = (const float*)d_in[0];   // [N,3] float32 logits
    const int*   yt  = (const int*)d_in[1];     // [N] labels in {-1,0,1}
    float*       out = (float*)d_out;           // scalar
    float*       ws  = (float*)d_ws;            // RED_BLOCKS partials
    const int n = in_sizes[1];

    mcc_loss_partials<<<RED_BLOCKS, RED_THREADS, 0, stream>>>(yp, yt, ws, n);
    mcc_loss_final_wmma<<<1, 32, 0, stream>>>(ws, out, RED_BLOCKS);
}